// GatedTestLayer_67370857005182
// MI455X (gfx1250) — compile-verified
//
#include <hip/hip_runtime.h>
#include <math.h>

#define DD 64
#define NN 50000
#define NE 800000
#define BN_EPS 1e-5f
#define AGG_EPS 1e-6f
#define SP 68   // padded LDS row stride (floats) for 16xD tiles
#define L2E 1.44269504088896340736f

typedef __attribute__((ext_vector_type(2))) float v2f;
typedef __attribute__((ext_vector_type(8))) float v8f;

__device__ __forceinline__ v8f wmma_f32(v2f a, v2f b, v8f c) {
  // D = A(16x4, f32) * B(4x16, f32) + C(16x16, f32)
  return __builtin_amdgcn_wmma_f32_16x16x4_f32(false, a, false, b, (short)0, c,
                                               false, false);
}

__device__ __forceinline__ float pclamp(float p) {
  return fminf(fmaxf(p, 1.0f), 100.0f);
}
__device__ __forceinline__ float powp(float x, float p) {
  // x >= 0 ; 0^p -> exp2(-inf) -> 0 matches |x|**p
  return exp2f(p * __log2f(x));
}

// ------------------------------------------------ async copy helpers (CDNA5)
// 16B global -> LDS, tracked by ASYNCcnt (GVS mode: saddr base + 32-bit voff).
// Low 32 bits of a generic shared pointer are the LDS byte offset (ISA 10.2).
__device__ __forceinline__ void async_cp16(void* lds_ptr, const void* gbase,
                                           unsigned voff_bytes) {
  unsigned lds_addr = (unsigned)(uintptr_t)lds_ptr;
  asm volatile("global_load_async_to_lds_b128 %0, %1, %2"
               :: "v"(lds_addr), "v"(voff_bytes), "s"(gbase)
               : "memory");
}
__device__ __forceinline__ void wait_async0() {
  asm volatile("s_wait_asynccnt 0x0" ::: "memory");
}
__device__ __forceinline__ void wait_async1() {
  asm volatile("s_wait_asynccnt 0x1" ::: "memory");
}
__device__ __forceinline__ void wait_async4() {
  asm volatile("s_wait_asynccnt 0x4" ::: "memory");
}

// ---------------------------------------------------------------- zero
__global__ void gg_zero(float* __restrict__ p, int n) {
  for (int i = blockIdx.x * blockDim.x + threadIdx.x; i < n;
       i += gridDim.x * blockDim.x)
    p[i] = 0.0f;
}

// ---------------------------------------------------------------- node GEMMs
// 8 waves/block: wave w -> weight (w>>1 in {A,B,D,E}), column half (w&1).
// Weights held in registers; h tiles double-buffered via async global->LDS.
__global__ __launch_bounds__(256) void gg_node_gemm(
    const float* __restrict__ h,
    const float* __restrict__ WA, const float* __restrict__ bA,
    const float* __restrict__ WB, const float* __restrict__ bB,
    const float* __restrict__ WD, const float* __restrict__ bD,
    const float* __restrict__ WE, const float* __restrict__ bE,
    const float* __restrict__ P,
    float* __restrict__ Ah, float* __restrict__ Bhp,
    float* __restrict__ Dh, float* __restrict__ Eh) {
  __shared__ float ldsH[2][16 * SP];

  const int tid = threadIdx.x;
  const int w = tid >> 5, lane = tid & 31, half = lane >> 4, lr = lane & 15;
  const int wg = w >> 1;           // 0:A 1:B 2:D 3:E
  const int nbase = (w & 1) * 32;  // column half

  const float* Wsel = (wg == 0) ? WA : (wg == 1) ? WB : (wg == 2) ? WD : WE;
  const float* bsel = (wg == 0) ? bA : (wg == 1) ? bB : (wg == 2) ? bD : bE;
  float* osel = (wg == 0) ? Ah : (wg == 1) ? Bhp : (wg == 2) ? Dh : Eh;

  // preload B operands: B[nt][k0] covers rows 4k0+2*half..+1, cols nbase+nt*16+lr
  v2f Breg[2][16];
#pragma unroll
  for (int k0 = 0; k0 < 16; ++k0) {
    const int r0 = 4 * k0 + 2 * half;
    const int c = nbase + lr;
    Breg[0][k0].x = Wsel[r0 * DD + c];
    Breg[0][k0].y = Wsel[(r0 + 1) * DD + c];
    Breg[1][k0].x = Wsel[r0 * DD + c + 16];
    Breg[1][k0].y = Wsel[(r0 + 1) * DD + c + 16];
  }
  float bias[2], pn[2];
#pragma unroll
  for (int nt = 0; nt < 2; ++nt) {
    const int n = nbase + nt * 16 + lr;
    bias[nt] = bsel[n];
    pn[nt] = pclamp(P[n]);
  }

  const int r = tid >> 4, c4 = tid & 15;  // this thread's float4 in the tile
  const int ntiles = NN / 16;             // 3125
  const int g = gridDim.x;
  const int t0 = blockIdx.x;
  if (t0 >= ntiles) return;

  // prologue: async-stage first tile into buffer 0
  async_cp16(&ldsH[0][r * SP + c4 * 4], h,
             (unsigned)(((t0 * 16 + r) * 16 + c4) * 16));

  int buf = 0;
  for (int tile = t0; tile < ntiles; tile += g, buf ^= 1) {
    const int tn = tile + g;
    if (tn < ntiles) {  // stage next tile into other buffer, overlap compute
      async_cp16(&ldsH[buf ^ 1][r * SP + c4 * 4], h,
                 (unsigned)(((tn * 16 + r) * 16 + c4) * 16));
      wait_async1();
    } else {
      wait_async0();
    }
    __syncthreads();

    v8f acc[2] = {};
#pragma unroll
    for (int k0 = 0; k0 < 16; ++k0) {
      v2f a;
      a.x = ldsH[buf][lr * SP + 4 * k0 + 2 * half];
      a.y = ldsH[buf][lr * SP + 4 * k0 + 2 * half + 1];
      acc[0] = wmma_f32(a, Breg[0][k0], acc[0]);
      acc[1] = wmma_f32(a, Breg[1][k0], acc[1]);
    }

#pragma unroll
    for (int nt = 0; nt < 2; ++nt) {
      const int n = nbase + nt * 16 + lr;
#pragma unroll
      for (int i = 0; i < 8; ++i) {
        const int m = i + 8 * half;
        float v = acc[nt][i] + bias[nt];
        if (wg == 1) v = powp(fabsf(v), pn[nt]);  // |Bh|^p
        osel[(tile * 16 + m) * DD + n] = v;
      }
    }
    __syncthreads();
  }
}

// ---------------------------------------------------------------- edge fused
// 8 waves/block: 4 edge tiles (16 edges each); 2 waves per tile (32 cols each).
// e tiles double-buffered via async global->LDS.
__global__ __launch_bounds__(256) void gg_edge(
    const float* __restrict__ e, const int* __restrict__ src,
    const int* __restrict__ dst, const float* __restrict__ WC,
    const float* __restrict__ bC, const float* __restrict__ P,
    const float* __restrict__ Dh, const float* __restrict__ Eh,
    const float* __restrict__ Bhp, float* __restrict__ e_pre,
    float* __restrict__ ssh, float* __restrict__ ss,
    float* __restrict__ bn_sum_e, float* __restrict__ bn_ssq_e) {
  __shared__ float ldsE[2][4][16 * SP];

  const int tid = threadIdx.x;
  const int w = tid >> 5, lane = tid & 31, half = lane >> 4, lr = lane & 15;
  const int tslot = w >> 1;        // which of the 4 tiles
  const int nbase = (w & 1) * 32;  // column half

  v2f Breg[2][16];
#pragma unroll
  for (int k0 = 0; k0 < 16; ++k0) {
    const int r0 = 4 * k0 + 2 * half;
    const int c = nbase + lr;
    Breg[0][k0].x = WC[r0 * DD + c];
    Breg[0][k0].y = WC[(r0 + 1) * DD + c];
    Breg[1][k0].x = WC[r0 * DD + c + 16];
    Breg[1][k0].y = WC[(r0 + 1) * DD + c + 16];
  }
  float bias[2], pn[2], bsum[2] = {0.f, 0.f}, bssq[2] = {0.f, 0.f};
#pragma unroll
  for (int nt = 0; nt < 2; ++nt) {
    const int n = nbase + nt * 16 + lr;
    bias[nt] = bC[n];
    pn[nt] = pclamp(P[n]);
  }

  const int nquads = NE / 64;  // 12500 groups of 4 tiles
  const int g = gridDim.x;
  const int q0 = blockIdx.x;
  if (q0 >= nquads) return;

  // prologue: stage quad q0 into buffer 0 (4 x 16B per thread)
#pragma unroll
  for (int it = 0; it < 4; ++it) {
    const int gi = tid + it * 256;
    const int slot = gi >> 8, fi = gi & 255;
    const int rr = fi >> 4, cc = fi & 15;
    async_cp16(&ldsE[0][slot][rr * SP + cc * 4], e,
               (unsigned)((((q0 * 4 + slot) * 16 + rr) * 16 + cc) * 16));
  }

  int buf = 0;
  for (int q = q0; q < nquads; q += g, buf ^= 1) {
    const int qn = q + g;
    if (qn < nquads) {
#pragma unroll
      for (int it = 0; it < 4; ++it) {
        const int gi = tid + it * 256;
        const int slot = gi >> 8, fi = gi & 255;
        const int rr = fi >> 4, cc = fi & 15;
        async_cp16(&ldsE[buf ^ 1][slot][rr * SP + cc * 4], e,
                   (unsigned)((((qn * 4 + slot) * 16 + rr) * 16 + cc) * 16));
      }
      wait_async4();  // the 4 older issues (current buffer) are complete
    } else {
      wait_async0();
    }
    __syncthreads();

    v8f acc[2] = {};
#pragma unroll
    for (int k0 = 0; k0 < 16; ++k0) {
      v2f a;
      a.x = ldsE[buf][tslot][lr * SP + 4 * k0 + 2 * half];
      a.y = ldsE[buf][tslot][lr * SP + 4 * k0 + 2 * half + 1];
      acc[0] = wmma_f32(a, Breg[0][k0], acc[0]);
      acc[1] = wmma_f32(a, Breg[1][k0], acc[1]);
    }

    const int ebase = (q * 4 + tslot) * 16;
#pragma unroll
    for (int i = 0; i < 8; ++i) {
      const int m = i + 8 * half;
      const int row = ebase + m;
      const int s = src[row];
      const int dd = dst[row];
#pragma unroll
      for (int nt = 0; nt < 2; ++nt) {
        const int n = nbase + nt * 16 + lr;
        const float ce = acc[nt][i] + bias[nt];
        const float ev = Dh[s * DD + n] + Eh[dd * DD + n] + ce;
        e_pre[row * DD + n] = ev;
        bsum[nt] += ev;
        bssq[nt] += ev * ev;
        const float sg = 1.0f / (1.0f + exp2f(-L2E * ev));  // sigmoid
        const float sp = powp(sg, pn[nt]);                  // sigma^p
        const float mv = Bhp[s * DD + n] * sp;
        atomicAdd(&ssh[dd * DD + n], mv);
        atomicAdd(&ss[dd * DD + n], sp);
      }
    }
    __syncthreads();
  }
#pragma unroll
  for (int nt = 0; nt < 2; ++nt) {
    const int n = nbase + nt * 16 + lr;
    atomicAdd(&bn_sum_e[n], bsum[nt]);
    atomicAdd(&bn_ssq_e[n], bssq[nt]);
  }
}

// ---------------------------------------------------------------- node finalize
__global__ __launch_bounds__(256) void gg_node_fin(
    const float* __restrict__ Ah, const float* __restrict__ ssh,
    const float* __restrict__ ss, const float* __restrict__ P,
    float* __restrict__ h_pre, float* __restrict__ bn_sum_h,
    float* __restrict__ bn_ssq_h) {
  const int tid = blockIdx.x * blockDim.x + threadIdx.x;
  const int col = tid & (DD - 1);
  const float ip = 1.0f / pclamp(P[col]);
  float lsum = 0.f, lssq = 0.f;
  const int stride = gridDim.x * blockDim.x;  // multiple of 64
  for (int i = tid; i < NN * DD; i += stride) {
    const float r = ssh[i] / (ss[i] + AGG_EPS);
    const float v = Ah[i] + exp2f(ip * __log2f(r));
    h_pre[i] = v;
    lsum += v;
    lssq += v * v;
  }
  atomicAdd(&bn_sum_h[col], lsum);
  atomicAdd(&bn_ssq_h[col], lssq);
}

// ---------------------------------------------------------------- BN stats
__global__ void gg_bn_fin(const float* __restrict__ bn, float* __restrict__ st) {
  const int c = threadIdx.x;
  if (c < DD) {
    const float mh = bn[c] / (float)NN;
    const float vh = bn[64 + c] / (float)NN - mh * mh;
    st[c] = mh;
    st[64 + c] = rsqrtf(vh + BN_EPS);
    const float me = bn[128 + c] / (float)NE;
    const float ve = bn[192 + c] / (float)NE - me * me;
    st[128 + c] = me;
    st[192 + c] = rsqrtf(ve + BN_EPS);
  }
}

// ---------------------------------------------------------------- final (BN+ReLU+res)
__global__ __launch_bounds__(256) void gg_final(
    const float* __restrict__ xin, const float* __restrict__ gamma,
    const float* __restrict__ beta, const float* __restrict__ st, int stoff,
    float* __restrict__ out, int n4) {
  const float4* in4 = reinterpret_cast<const float4*>(xin);
  float4* o4 = reinterpret_cast<float4*>(out);
  for (int i = blockIdx.x * blockDim.x + threadIdx.x; i < n4;
       i += gridDim.x * blockDim.x) {
    const int c0 = (i & 15) * 4;
    float4 x = o4[i];
    float4 r = in4[i];
    float* xp = &x.x;
    float* rp = &r.x;
    float4 y;
    float* yp = &y.x;
#pragma unroll
    for (int j = 0; j < 4; ++j) {
      const int c = c0 + j;
      float v = gamma[c] * (xp[j] - st[stoff + c]) * st[stoff + 64 + c] + beta[c];
      v = fmaxf(v, 0.0f);
      yp[j] = rp[j] + v;
    }
    o4[i] = y;
  }
}

// ---------------------------------------------------------------- launch
extern "C" void kernel_launch(void* const* d_in, const int* in_sizes, int n_in,
                              void* d_out, int out_size, void* d_ws,
                              size_t ws_size, hipStream_t stream) {
  const float* h = (const float*)d_in[0];
  const float* e = (const float*)d_in[1];
  const int* src = (const int*)d_in[2];
  const int* dst = (const int*)d_in[3];
  const float* WA = (const float*)d_in[4];
  const float* bA = (const float*)d_in[5];
  const float* WB = (const float*)d_in[6];
  const float* bB = (const float*)d_in[7];
  const float* WC = (const float*)d_in[8];
  const float* bC = (const float*)d_in[9];
  const float* WD = (const float*)d_in[10];
  const float* bD = (const float*)d_in[11];
  const float* WE = (const float*)d_in[12];
  const float* bE = (const float*)d_in[13];
  const float* P = (const float*)d_in[14];
  const float* gamma_h = (const float*)d_in[15];
  const float* beta_h = (const float*)d_in[16];
  const float* gamma_e = (const float*)d_in[17];
  const float* beta_e = (const float*)d_in[18];

  float* out = (float*)d_out;             // [h_out (NN*64) | e_out (NE*64)]
  float* h_pre = out;                     // pre-BN h lives in-place
  float* e_pre = out + (size_t)NN * DD;   // pre-BN e lives in-place

  float* ws = (float*)d_ws;
  float* Ah = ws;                         // NN*64
  float* Bhp = ws + 3200000;              // NN*64
  float* Dh = ws + 6400000;               // NN*64
  float* Eh = ws + 9600000;               // NN*64
  float* ssh = ws + 12800000;             // NN*64   (sum_sigma_h)
  float* ss = ws + 16000000;              // NN*64   (sum_sigma)
  float* bn = ws + 19200000;              // 256: sum_h, ssq_h, sum_e, ssq_e
  float* st = bn + 256;                   // 256: mean_h, rstd_h, mean_e, rstd_e

  // zero accumulators (ssh, ss, bn, st are contiguous)
  gg_zero<<<2048, 256, 0, stream>>>(ssh, 6400512);

  gg_node_gemm<<<1024, 256, 0, stream>>>(h, WA, bA, WB, bB, WD, bD, WE, bE, P,
                                         Ah, Bhp, Dh, Eh);

  gg_edge<<<2048, 256, 0, stream>>>(e, src, dst, WC, bC, P, Dh, Eh, Bhp, e_pre,
                                    ssh, ss, bn + 128, bn + 192);

  gg_node_fin<<<512, 256, 0, stream>>>(Ah, ssh, ss, P, h_pre, bn, bn + 64);

  gg_bn_fin<<<1, 64, 0, stream>>>(bn, st);

  gg_final<<<1024, 256, 0, stream>>>(h, gamma_h, beta_h, st, 0, h_pre,
                                     NN * DD / 4);
  gg_final<<<2048, 256, 0, stream>>>(e, gamma_e, beta_e, st, 128, e_pre,
                                     NE * DD / 4);
}